// ExpertsChooseExpand_25348896981195
// MI455X (gfx1250) — compile-verified
//
#include <hip/hip_runtime.h>

typedef float v2f __attribute__((ext_vector_type(2)));
typedef float v8f __attribute__((ext_vector_type(8)));

// async-to-LDS builtin takes pointers to 16-byte int vectors
typedef int v4i_vs __attribute__((vector_size(16)));
typedef __attribute__((address_space(1))) v4i_vs* gv4p;
typedef __attribute__((address_space(3))) v4i_vs* lv4p;

#if defined(__has_builtin)
#if __has_builtin(__builtin_amdgcn_global_load_async_to_lds_b128)
#define HAVE_ASYNC_LDS 1
#endif
#endif

#define B_ 8
#define E_ 8
#define C_ 1024
#define I_ 128
#define O_ 256
#define T_ 8192

#define TILE_C 128
#define LDS_K  132   // padded row stride (floats): 132 % 64 banks = 4 -> conflict-free b64 reads

__global__ void moe_zero_out(float4* __restrict__ out, int n4) {
    int idx = blockIdx.x * blockDim.x + threadIdx.x;
    int stride = gridDim.x * blockDim.x;
    float4 z = make_float4(0.f, 0.f, 0.f, 0.f);
    for (int i = idx; i < n4; i += stride) out[i] = z;
}

// Stage a 128-row x 128-float tile (row-major, contiguous) into LDS with
// padded row stride LDS_K.  256 threads x 16 float4 each.
__device__ __forceinline__ void stage_tile(const float* __restrict__ gsrc,
                                           float* lds_base, int tid) {
#ifdef HAVE_ASYNC_LDS
    #pragma unroll
    for (int i = 0; i < 16; ++i) {
        int f   = tid + i * 256;     // 4096 float4 total
        int row = f >> 5;            // 32 float4 per row
        int kf  = f & 31;
        __builtin_amdgcn_global_load_async_to_lds_b128(
            (gv4p)(gsrc + (size_t)f * 4),
            (lv4p)(&lds_base[row * LDS_K + kf * 4]),
            0, 0);
    }
#else
    const float4* src = (const float4*)gsrc;
    #pragma unroll
    for (int i = 0; i < 16; ++i) {
        int f   = tid + i * 256;
        int row = f >> 5;
        int kf  = f & 31;
        float4 v = src[f];
        *(float4*)(&lds_base[row * LDS_K + kf * 4]) = v;
    }
#endif
}

__device__ __forceinline__ void wait_stage_done() {
#ifdef HAVE_ASYNC_LDS
#if __has_builtin(__builtin_amdgcn_s_wait_asynccnt)
    __builtin_amdgcn_s_wait_asynccnt(0);
#else
    asm volatile("s_wait_asynccnt 0" ::: "memory");
#endif
#endif
    __syncthreads();
}

__global__ __launch_bounds__(256) void moe_gemm_scatter(
    const float* __restrict__ x,     // (B,E,C,I)
    const int*   __restrict__ eidx,  // (B,E,C)
    const float* __restrict__ gate,  // (B,E,C)
    const float* __restrict__ w,     // (E,O,I)
    const float* __restrict__ bias,  // (O)
    float*       __restrict__ out)   // (B,T,O)
{
    __shared__ float Xs[TILE_C * LDS_K];   // 128 x 132 floats = 66 KB
    __shared__ float Ws[128 * LDS_K];      // half of W^T:  128 x 132 = 66 KB

    const int ct   = blockIdx.x;
    const int e    = blockIdx.y;
    const int b    = blockIdx.z;
    const int tid  = threadIdx.x;
    const int wave = tid >> 5;
    const int lane = tid & 31;
    const int lhalf = lane >> 4;   // 0: K+0/1 (or M 0..7), 1: K+2/3 (or M 8..15)
    const int l16   = lane & 15;

    const size_t be = (size_t)b * E_ + e;
    const int c0 = ct * TILE_C;

    // ---- stage X tile: 128 rows x 128 floats (async global->LDS) ----
    stage_tile(x + (be * C_ + c0) * (size_t)I_, Xs, tid);

    // ---- per-lane gate / index cache for this wave's 16 rows ----
    // C/D layout: VGPR r holds M=r (lanes 0-15) and M=8+r (lanes 16-31)
    float gates[8];
    int   idxs[8];
    {
        const size_t gbase = be * C_ + (size_t)c0 + wave * 16 + lhalf * 8;
        #pragma unroll
        for (int r = 0; r < 8; ++r) {
            gates[r] = gate[gbase + r];
            idxs[r]  = eidx[gbase + r];
        }
    }

    v8f acc[8];

    for (int h = 0; h < 2; ++h) {
        // ---- stage half of W: rows h*128 .. h*128+127, each I_=128 floats ----
        stage_tile(w + ((size_t)e * O_ + h * 128) * I_, Ws, tid);
        wait_stage_done();   // s_wait_asynccnt 0 + barrier (covers Xs for h==0)

        #pragma unroll
        for (int nt = 0; nt < 8; ++nt) acc[nt] = (v8f){0,0,0,0,0,0,0,0};

        // A fragment: row = wave*16 + l16, K = k0 + 2*lhalf .. +1
        const float* xrow = &Xs[(wave * 16 + l16) * LDS_K + lhalf * 2];

        #pragma unroll 4
        for (int k0 = 0; k0 < I_; k0 += 4) {
            v2f a = *(const v2f*)(xrow + k0);
            #pragma unroll
            for (int nt = 0; nt < 8; ++nt) {
                // B fragment: B[k][n] = W[n][k]; n = nt*16 + l16
                v2f bf = *(const v2f*)(&Ws[(nt * 16 + l16) * LDS_K + k0 + lhalf * 2]);
                acc[nt] = __builtin_amdgcn_wmma_f32_16x16x4_f32(
                    false, a, false, bf, (short)0, acc[nt], false, false);
            }
        }

        // ---- epilogue for this half: (acc + bias) * gate -> atomic scatter ----
        #pragma unroll
        for (int nt = 0; nt < 8; ++nt) {
            int col = h * 128 + nt * 16 + l16;
            float bv = bias[col];
            #pragma unroll
            for (int r = 0; r < 8; ++r) {
                float val = (acc[nt][r] + bv) * gates[r];
                float* dst = out + ((size_t)b * T_ + idxs[r]) * O_ + col;
                atomicAdd(dst, val);
            }
        }
        __syncthreads();   // all waves done reading Ws before restage
    }
}

extern "C" void kernel_launch(void* const* d_in, const int* in_sizes, int n_in,
                              void* d_out, int out_size, void* d_ws, size_t ws_size,
                              hipStream_t stream) {
    const float* x    = (const float*)d_in[0];
    const int*   eidx = (const int*)  d_in[1];
    const float* gate = (const float*)d_in[2];
    const float* w    = (const float*)d_in[3];
    const float* bias = (const float*)d_in[4];
    float* out = (float*)d_out;

    int n4 = (B_ * T_ * O_) / 4;   // 4,194,304 float4
    moe_zero_out<<<2048, 256, 0, stream>>>((float4*)out, n4);

    dim3 grid(C_ / TILE_C, E_, B_);
    moe_gemm_scatter<<<grid, 256, 0, stream>>>(x, eidx, gate, w, bias, out);
}